// SS2D_32272384262444
// MI455X (gfx1250) — compile-verified
//
#include <hip/hip_runtime.h>
#include <hip/hip_bf16.h>
#include <math.h>

// ---------------- problem constants ----------------
constexpr int DM  = 128;          // d_model
constexpr int DI  = 256;          // d_inner
constexpr int NST = 16;           // d_state
constexpr int RNK = 8;            // dt_rank
constexpr int KD  = 4;            // scan directions
constexpr int BB  = 4;
constexpr int HH  = 56;
constexpr int WW  = 56;
constexpr int LL  = HH * WW;      // 3136
constexpr int XDP = 48;           // padded x_dbl rows (40 used: 8 dt, 16 B, 16 C)
constexpr int NTJ = 4;            // N-tiles per wave in GEMMs (16x64 output/wave)

typedef float v2f __attribute__((ext_vector_type(2)));
typedef float v8f __attribute__((ext_vector_type(8)));

// ---------------- f32 WMMA helper (16x16x4, D = A*B + C) ----------------
__device__ __forceinline__ v8f wmma_f32(v2f a, v2f b, v8f c) {
#if defined(__HIP_DEVICE_COMPILE__)
    // 8 args: (neg_a, A, neg_b, B, c_mod, C, reuse_a, reuse_b)
    return __builtin_amdgcn_wmma_f32_16x16x4_f32(false, a, false, b, (short)0, c, false, false);
#else
    (void)a; (void)b;
    return c;
#endif
}

// fast reciprocal: denominator of silu is in [1, inf) -> v_rcp_f32 is safe,
// avoids the ~10-op IEEE divide expansion seen in round-1 asm.
__device__ __forceinline__ float fast_rcp(float x) {
#if defined(__HIP_DEVICE_COMPILE__)
    return __builtin_amdgcn_rcpf(x);
#else
    return 1.0f / x;
#endif
}

__device__ __forceinline__ float silu_f(float x) {
    return x * fast_rcp(1.0f + __expf(-x));
}

__device__ __forceinline__ void prefetch_g(const void* p) {
#if defined(__HIP_DEVICE_COMPILE__)
    __builtin_prefetch(p, 0, 0);      // -> global_prefetch_b8
#else
    (void)p;
#endif
}

// scan order k, position l  ->  spatial flat index. Involution for H==W,
// so it is simultaneously the cross_scan gather map and cross_merge scatter map.
__device__ __forceinline__ int scan_src(int k, int l) {
    int ll = (k >= 2) ? (LL - 1 - l) : l;
    if (k & 1) {                      // transpose order: l = w*H + h reads h*W + w
        int wv = ll / HH;
        int hv = ll % HH;
        ll = hv * WW + wv;
    }
    return ll;
}

// =======================================================================
// 1) in_proj: xz = W(512x128) @ x_b(128xL). rows<256 -> xx_raw, rows>=256 -> silu -> z
//    one wave per 16x64 tile: A fragment loaded once per K-step, reused 4x.
// =======================================================================
__global__ void __launch_bounds__(32)
in_proj_kernel(const float* __restrict__ W, const float* __restrict__ X,
               float* __restrict__ xx_raw, float* __restrict__ zs) {
    const int lane = threadIdx.x & 31;
    const int half = lane >> 4;        // 0: lanes 0-15, 1: lanes 16-31
    const int mr   = lane & 15;
    const int m0   = blockIdx.x * 16;  // 0..511
    const int n0   = blockIdx.y * (16 * NTJ);
    const int b    = blockIdx.z;
    const float* Xb = X + (size_t)b * DM * LL;

    v8f acc[NTJ];
    #pragma unroll
    for (int j = 0; j < NTJ; ++j) acc[j] = v8f{0.f,0.f,0.f,0.f,0.f,0.f,0.f,0.f};

    for (int kb = 0; kb < DM; kb += 4) {
        v2f a;
        // A 16x4 layout: lanes0-15 hold K=kb,kb+1 ; lanes16-31 hold K=kb+2,kb+3
        a.x = W[(m0 + mr) * DM + kb + 2 * half + 0];
        a.y = W[(m0 + mr) * DM + kb + 2 * half + 1];
        #pragma unroll
        for (int j = 0; j < NTJ; ++j) {
            const int col = n0 + 16 * j + mr;
            v2f bf;
            // B 4x16: VGPR0 rows {kb, kb+1} across halves, VGPR1 rows {kb+2, kb+3}
            bf.x = Xb[(size_t)(kb + half) * LL + col];
            bf.y = Xb[(size_t)(kb + 2 + half) * LL + col];
            acc[j] = wmma_f32(a, bf, acc[j]);
        }
    }
    #pragma unroll
    for (int j = 0; j < NTJ; ++j) {
        const int col = n0 + 16 * j + mr;
        #pragma unroll
        for (int v = 0; v < 8; ++v) {
            int row = m0 + v + 8 * half;   // C/D: VGPR v -> rows v / v+8
            float val = acc[j][v];
            if (row < DI) {
                xx_raw[((size_t)b * DI + row) * LL + col] = val;
            } else {
                zs[((size_t)b * DI + (row - DI)) * LL + col] = silu_f(val);
            }
        }
    }
}

// =======================================================================
// 2) depthwise 3x3 conv (SAME) + bias + SiLU
// =======================================================================
__global__ void __launch_bounds__(256)
conv_silu_kernel(const float* __restrict__ xx_raw, const float* __restrict__ cw,
                 const float* __restrict__ cb, float* __restrict__ xxc) {
    int idx = blockIdx.x * blockDim.x + threadIdx.x;      // ((b*DI+d)*LL + l)
    if (idx >= BB * DI * LL) return;
    int l = idx % LL;
    int d = (idx / LL) % DI;
    int h = l / WW, w = l % WW;
    const float* src = xx_raw + (size_t)(idx - l);        // (b,d) plane
    const float* wk  = cw + d * 9;
    float s = cb[d];
    #pragma unroll
    for (int kh = -1; kh <= 1; ++kh) {
        int hh = h + kh;
        if (hh < 0 || hh >= HH) continue;
        #pragma unroll
        for (int kw = -1; kw <= 1; ++kw) {
            int ww2 = w + kw;
            if (ww2 < 0 || ww2 >= WW) continue;
            s += wk[(kh + 1) * 3 + (kw + 1)] * src[hh * WW + ww2];
        }
    }
    xxc[idx] = silu_f(s);
}

// =======================================================================
// 3) x_proj: x_dbl(b,k) = x_proj_w[k](40x256) @ xs(b,k)(256xL)
//    cross-scan gather fused into the B-operand address; M padded 40->48.
//    16x64 per wave; output stored l-major (48-float rows) for the scan.
// =======================================================================
__global__ void __launch_bounds__(32)
x_proj_kernel(const float* __restrict__ xpw, const float* __restrict__ xxc,
              float* __restrict__ xdbl) {
    const int lane = threadIdx.x & 31;
    const int half = lane >> 4;
    const int mr   = lane & 15;
    const int m0   = blockIdx.x * 16;              // 0,16,32  (rows 40..47 are pad)
    const int n0   = blockIdx.y * (16 * NTJ);
    const int bk   = blockIdx.z;                   // b*4 + k
    const int k    = bk & 3;
    const int b    = bk >> 2;

    int src[NTJ];                                  // fused cross_scan gather
    #pragma unroll
    for (int j = 0; j < NTJ; ++j) src[j] = scan_src(k, n0 + 16 * j + mr);

    const float* Ak = xpw + (size_t)k * 40 * DI;
    const float* Xb = xxc + (size_t)b * DI * LL;
    const int arow = m0 + mr;
    const bool av  = (arow < 40);

    v8f acc[NTJ];
    #pragma unroll
    for (int j = 0; j < NTJ; ++j) acc[j] = v8f{0.f,0.f,0.f,0.f,0.f,0.f,0.f,0.f};

    for (int kb = 0; kb < DI; kb += 4) {
        v2f a;
        a.x = av ? Ak[arow * DI + kb + 2 * half + 0] : 0.f;
        a.y = av ? Ak[arow * DI + kb + 2 * half + 1] : 0.f;
        #pragma unroll
        for (int j = 0; j < NTJ; ++j) {
            v2f bf;
            bf.x = Xb[(size_t)(kb + half) * LL + src[j]];
            bf.y = Xb[(size_t)(kb + 2 + half) * LL + src[j]];
            acc[j] = wmma_f32(a, bf, acc[j]);
        }
    }
    #pragma unroll
    for (int j = 0; j < NTJ; ++j) {
        float* outl = xdbl + ((size_t)bk * LL + (n0 + 16 * j + mr)) * XDP;
        #pragma unroll
        for (int v = 0; v < 8; ++v) {
            int row = m0 + v + 8 * half;
            if (row < 40) outl[row] = acc[j][v];
        }
    }
}

// =======================================================================
// 4) selective scan, parallel over (b,k,d,n): one state element per lane.
//    dt_proj (rank 8) + softplus computed inline (broadcast loads).
//    einsum over n done with wave32 width-16 xor-shuffle reduction.
//    Output written directly to its cross_merge position (scan_src is the
//    same map), so merge = plain 4-way sum later. No atomics -> deterministic.
// =======================================================================
__global__ void __launch_bounds__(256)
scan_kernel(const float* __restrict__ xxc,   const float* __restrict__ xdbl,
            const float* __restrict__ dtw_all, const float* __restrict__ dt_bias,
            const float* __restrict__ A_logs,  const float* __restrict__ Ds,
            float* __restrict__ ys) {
    int t = blockIdx.x * blockDim.x + threadIdx.x;
    if (t >= BB * KD * DI * NST) return;
    const int n    = t & 15;
    const int rest = t >> 4;
    const int d    = rest % DI;
    const int k    = (rest / DI) % KD;
    const int b    = rest / (DI * KD);
    const int kd   = k * DI + d;

    const float An   = -__expf(A_logs[kd * NST + n]);   // A = -exp(A_log)
    const float bias = dt_bias[kd];
    const float Dsk  = Ds[kd];
    float dtw[RNK];
    #pragma unroll
    for (int r = 0; r < RNK; ++r) dtw[r] = dtw_all[kd * RNK + r];

    const float* u_base  = xxc  + ((size_t)b * DI + d) * LL;
    const float* xd_base = xdbl + (size_t)(b * KD + k) * LL * XDP;
    float* y_base        = ys   + ((size_t)(b * KD + k) * DI + d) * LL;

    float h = 0.f;
    for (int l = 0; l < LL; ++l) {
        const int src = scan_src(k, l);
        const float u = u_base[src];                       // broadcast in half-wave
        const float* xd = xd_base + (size_t)l * XDP;
        prefetch_g(xd + XDP);                              // next step's row
        float dts = bias;
        #pragma unroll
        for (int r = 0; r < RNK; ++r) dts += dtw[r] * xd[r];   // inline dt_proj
        const float delta = (dts > 20.f) ? dts : log1pf(__expf(dts)); // softplus
        const float Bn = xd[RNK + n];                      // contiguous per lane
        const float Cn = xd[RNK + NST + n];
        h = __expf(delta * An) * h + (delta * u) * Bn;     // state recurrence
        float part = h * Cn;                               // y = sum_n h_n*C_n
        part += __shfl_xor(part, 1, 16);
        part += __shfl_xor(part, 2, 16);
        part += __shfl_xor(part, 4, 16);
        part += __shfl_xor(part, 8, 16);
        if (n == 0) y_base[src] = part + u * Dsk;          // write at merge pos
    }
}

// =======================================================================
// 5) cross_merge (4-way sum) + LayerNorm over d + gamma/beta + * silu(z)
// =======================================================================
__global__ void __launch_bounds__(256)
ln_mul_kernel(const float* __restrict__ ys, const float* __restrict__ zs,
              const float* __restrict__ lnw, const float* __restrict__ lnb,
              float* __restrict__ yf) {
    const int bl = blockIdx.x;          // b*LL + l
    const int b  = bl / LL;
    const int l  = bl % LL;
    const int d  = threadIdx.x;         // 256 = DI

    float v = 0.f;
    #pragma unroll
    for (int k = 0; k < KD; ++k)
        v += ys[(((size_t)(b * KD + k)) * DI + d) * LL + l];

    __shared__ float red[DI];
    red[d] = v;
    __syncthreads();
    for (int s = DI / 2; s > 0; s >>= 1) {
        if (d < s) red[d] += red[d + s];
        __syncthreads();
    }
    const float mu = red[0] * (1.0f / DI);
    __syncthreads();
    const float dv = v - mu;
    red[d] = dv * dv;
    __syncthreads();
    for (int s = DI / 2; s > 0; s >>= 1) {
        if (d < s) red[d] += red[d + s];
        __syncthreads();
    }
    const float rs = rsqrtf(red[0] * (1.0f / DI) + 1e-5f);
    const float out = (dv * rs * lnw[d] + lnb[d]) * zs[((size_t)b * DI + d) * LL + l];
    yf[((size_t)b * DI + d) * LL + l] = out;
}

// =======================================================================
// 6) out_proj: out = W(128x256) @ y_b(256xL), 16x64 per wave
// =======================================================================
__global__ void __launch_bounds__(32)
out_proj_kernel(const float* __restrict__ W, const float* __restrict__ Y,
                float* __restrict__ out) {
    const int lane = threadIdx.x & 31;
    const int half = lane >> 4;
    const int mr   = lane & 15;
    const int m0   = blockIdx.x * 16;   // 0..127
    const int n0   = blockIdx.y * (16 * NTJ);
    const int b    = blockIdx.z;
    const float* Yb = Y + (size_t)b * DI * LL;

    v8f acc[NTJ];
    #pragma unroll
    for (int j = 0; j < NTJ; ++j) acc[j] = v8f{0.f,0.f,0.f,0.f,0.f,0.f,0.f,0.f};

    for (int kb = 0; kb < DI; kb += 4) {
        v2f a;
        a.x = W[(m0 + mr) * DI + kb + 2 * half + 0];
        a.y = W[(m0 + mr) * DI + kb + 2 * half + 1];
        #pragma unroll
        for (int j = 0; j < NTJ; ++j) {
            const int col = n0 + 16 * j + mr;
            v2f bf;
            bf.x = Yb[(size_t)(kb + half) * LL + col];
            bf.y = Yb[(size_t)(kb + 2 + half) * LL + col];
            acc[j] = wmma_f32(a, bf, acc[j]);
        }
    }
    #pragma unroll
    for (int j = 0; j < NTJ; ++j) {
        const int col = n0 + 16 * j + mr;
        #pragma unroll
        for (int v = 0; v < 8; ++v) {
            int row = m0 + v + 8 * half;
            out[((size_t)b * DM + row) * LL + col] = acc[j][v];
        }
    }
}

// =======================================================================
// launch
// =======================================================================
extern "C" void kernel_launch(void* const* d_in, const int* in_sizes, int n_in,
                              void* d_out, int out_size, void* d_ws, size_t ws_size,
                              hipStream_t stream) {
    (void)in_sizes; (void)n_in; (void)out_size; (void)ws_size;
    const float* x         = (const float*)d_in[0];   // (B,128,H,W)
    const float* in_proj_w = (const float*)d_in[1];   // (512,128)
    const float* conv2d_w  = (const float*)d_in[2];   // (256,1,3,3)
    const float* conv2d_b  = (const float*)d_in[3];   // (256)
    const float* x_proj_w  = (const float*)d_in[4];   // (4,40,256)
    const float* dt_proj_w = (const float*)d_in[5];   // (4,256,8)
    const float* dt_bias   = (const float*)d_in[6];   // (4,256)
    const float* A_logs    = (const float*)d_in[7];   // (1024,16)
    const float* Ds        = (const float*)d_in[8];   // (1024)
    const float* ln_w      = (const float*)d_in[9];   // (256)
    const float* ln_b      = (const float*)d_in[10];  // (256)
    const float* out_proj_w= (const float*)d_in[11];  // (128,256)
    float* out             = (float*)d_out;           // (B,128,H,W)

    // workspace carve-up (floats). ~100 MB total; yfin reuses xx_raw.
    float* ws = (float*)d_ws;
    const size_t planeBDL = (size_t)BB * DI * LL;                 // 3,211,264
    float* xx_raw = ws;                 ws += planeBDL;           // pre-conv xx (reused as yfin)
    float* zs     = ws;                 ws += planeBDL;           // silu(z)
    float* xxc    = ws;                 ws += planeBDL;           // post conv+silu
    float* xdbl   = ws;                 ws += (size_t)BB * KD * LL * XDP;  // 2,408,448
    float* ys     = ws;                 ws += (size_t)BB * KD * DI * LL;   // 12,845,056
    float* yfin   = xx_raw;             // reuse: xx_raw dead after conv kernel

    // 1) in_proj (WMMA f32, 16x64 tiles)
    in_proj_kernel<<<dim3(2 * DI / 16, LL / (16 * NTJ), BB), 32, 0, stream>>>(
        in_proj_w, x, xx_raw, zs);

    // 2) depthwise conv + silu
    {
        int n = BB * DI * LL;
        conv_silu_kernel<<<(n + 255) / 256, 256, 0, stream>>>(xx_raw, conv2d_w, conv2d_b, xxc);
    }

    // 3) x_proj (WMMA f32, fused cross-scan gather, 16x64 tiles)
    x_proj_kernel<<<dim3(3, LL / (16 * NTJ), BB * KD), 32, 0, stream>>>(x_proj_w, xxc, xdbl);

    // 4) selective scan (65536 lanes, one per (b,k,d,n))
    {
        int n = BB * KD * DI * NST;
        scan_kernel<<<(n + 255) / 256, 256, 0, stream>>>(
            xxc, xdbl, dt_proj_w, dt_bias, A_logs, Ds, ys);
    }

    // 5) merge + layernorm + *silu(z)
    ln_mul_kernel<<<BB * LL, DI, 0, stream>>>(ys, zs, ln_w, ln_b, yfin);

    // 6) out_proj (WMMA f32, 16x64 tiles)
    out_proj_kernel<<<dim3(DM / 16, LL / (16 * NTJ), BB), 32, 0, stream>>>(out_proj_w, yfin, out);
}